// CNS_residuals_75806172774705
// MI455X (gfx1250) — compile-verified
//
#include <hip/hip_runtime.h>
#include <cstdint>
#include <cstddef>

// ---------------------------------------------------------------------------
// CNS residual stencil, gfx1250 (MI455X).
// Memory-bound fp32 stencil -> no WMMA. Uses both CDNA5 data-movement paths:
//   * TENSOR_LOAD_TO_LDS (TDM, TENSORcnt) for rectangular tile DMA
//   * global_load_async_to_lds_b32 (ASYNCcnt) for predicated/zero-padded halos
// ---------------------------------------------------------------------------

#define NT 256          // 8 wave32 waves
#define TX 32           // tile size in x (output points)
#define TY 32           // tile size in y (output points)
#define HXY 36          // TX+4 halo-2 tile edge
#define NHALO (HXY * HXY)       // 1296
#define NCTR  (TX * TY)         // 1024

#define OFF_RH 0
#define OFF_UH (NHALO)
#define OFF_VH (2 * NHALO)
#define OFF_PH (3 * NHALO)
#define OFF_CT (4 * NHALO)
#define LDS_FLOATS (4 * NHALO + 6 * NCTR)   // 11328 floats = 45.3 KB

typedef uint32_t u32x4 __attribute__((ext_vector_type(4)));
typedef uint32_t u32x8 __attribute__((ext_vector_type(8)));

// Async copy: 4 bytes from global address g -> LDS byte offset lds_byte.
// GV mode: VDST = LDS addr VGPR, VADDR = 64-bit global address VGPR pair.
__device__ __forceinline__ void async_ld(uint32_t lds_byte, const float* g) {
    asm volatile("global_load_async_to_lds_b32 %0, %1, off"
                 :
                 : "v"(lds_byte), "v"((uint64_t)(uintptr_t)g)
                 : "memory");
}

__device__ __forceinline__ void wait_async0() {
    asm volatile("s_wait_asynccnt 0" ::: "memory");
}

// TDM: DMA a 2D f32 subrectangle (w elems/row, h rows, row stride in elems)
// from global memory to a contiguous LDS region (packed at pitch w).
// D# group0 (4 SGPRs) + group1 (8 SGPRs); groups 2/3 omitted (2D tensor).
__device__ __forceinline__ void tdm_load_2d(uint32_t lds_byte, const float* g,
                                            uint32_t w, uint32_t h,
                                            uint32_t stride_elems) {
    const uint64_t ga = (uint64_t)(uintptr_t)g;
    const uint32_t TD = 1u << 18;  // huge tensor_dim: disable OOB clamping
    u32x4 g0;
    g0[0] = 1u;                                   // count=1 (valid), user mode
    g0[1] = lds_byte;                             // lds_addr (bytes)
    g0[2] = (uint32_t)ga;                         // global_addr[31:0]
    g0[3] = ((uint32_t)(ga >> 32) & 0x01FFFFFFu)  // global_addr[56:32]
            | 0x80000000u;                        // type=2 ("image")
    u32x8 g1;
    g1[0] = 2u << 16;                             // wg_mask=0, data_size=4B
    g1[1] = (TD & 0xFFFFu) << 16;                 // barrier_addr=0, dim0[15:0]
    g1[2] = (TD >> 16) | ((TD & 0xFFFFu) << 16);  // dim0[31:16], dim1[15:0]
    g1[3] = (TD >> 16) | (w << 16);               // dim1[31:16], tile_dim0=w
    g1[4] = h;                                    // tile_dim1=h, tile_dim2=0
    g1[5] = stride_elems * 4u / 4u;               // dim0_stride[31:0] (elems)
    g1[6] = 0u;                                   // dim0_stride[47:32], dim1_stride lo
    g1[7] = 0u;                                   // dim1_stride hi
    asm volatile("tensor_load_to_lds %0, %1"
                 :
                 : "s"(g0), "s"(g1)
                 : "memory");
}

__global__ __launch_bounds__(NT) void cns_residual_kernel(
    const float* __restrict__ vars,
    const float* __restrict__ pdx, const float* __restrict__ pdt,
    const float* __restrict__ peta, const float* __restrict__ pzeta,
    float* __restrict__ out)
{
    __shared__ float smem[LDS_FLOATS];
    // Flat address of an LDS object carries the LDS byte offset in its low
    // 32 bits (generic ptr = {shared aperture hi, AS3 offset}).
    const uint32_t lds_base = (uint32_t)(uintptr_t)(void*)&smem[0];

    const int b = (int)blockIdx.z;
    const int t = (int)blockIdx.y + 1;               // t in [1, 62]
    const int tile = (int)blockIdx.x;                // 16 tiles: 4 in x, 4 in y
    const int tlx = tile & 3;
    const int tly = tile >> 2;
    const int x_base = 1 + tlx * TX;                 // {1,33,65,97}
    const int y_base = 1 + tly * TY;                 // {1,33,65,97}
    const int tid = (int)threadIdx.x;
    // Fully interior tiles: whole 36x36 halo rectangle lies inside [0,128)^2.
    const bool interior = (tlx == 1 || tlx == 2) && (tly == 1 || tly == 2);

    const size_t CST = 64ull * 128ull * 128ull;      // channel stride (elems)
    const float* bb  = vars + (size_t)b * 4ull * CST;
    const float* pt  = bb + (size_t)t * 16384ull;    // time-t plane, channel 0
    const float* ptm = pt - 16384;                   // t-1
    const float* ptp = pt + 16384;                   // t+1

    const float fdx = *pdx, fdt = *pdt, feta = *peta, fzeta = *pzeta;

    // ---- Phase 1a: TDM DMA (wave 0 only; EXEC all-ones inside) ----
    if (threadIdx.x < 32) {
        const size_t go0 = ((size_t)x_base << 7) + (size_t)y_base;
        // t+-1 center tiles: always 32x32; far-edge overreads stay inside the
        // vars buffer and feed only masked-out outputs.
#pragma unroll
        for (int c = 0; c < 3; ++c) {
            tdm_load_2d(lds_base + (uint32_t)((OFF_CT + c * NCTR) * 4),
                        ptm + (size_t)c * CST + go0, TY, TX, 128u);
            tdm_load_2d(lds_base + (uint32_t)((OFF_CT + (3 + c) * NCTR) * 4),
                        ptp + (size_t)c * CST + go0, TY, TX, 128u);
        }
        if (interior) {
            // Whole halo-2 rectangle in-domain: DMA all four t-plane fields.
            const size_t goH = ((size_t)(x_base - 2) << 7) + (size_t)(y_base - 2);
#pragma unroll
            for (int c = 0; c < 4; ++c) {
                tdm_load_2d(lds_base + (uint32_t)(c * NHALO * 4),
                            pt + (size_t)c * CST + goH, HXY, HXY, 128u);
            }
        }
    }

    // ---- Phase 1b: boundary tiles: zero ring + predicated async loads ----
    if (!interior) {
        for (int i = tid; i < 4 * NHALO; i += NT) smem[i] = 0.0f;
    }
    __syncthreads();
    if (!interior) {
        for (int i = tid; i < NHALO; i += NT) {
            const int hx = i / HXY;
            const int hy = i - hx * HXY;
            const int gx = x_base - 2 + hx;
            const int gy = y_base - 2 + hy;
            if ((unsigned)gx < 128u && (unsigned)gy < 128u) {
                const size_t go = ((size_t)gx << 7) + (size_t)gy;
                const uint32_t lb = lds_base + (uint32_t)(i * 4);
                async_ld(lb + OFF_RH * 4, pt + go);
                async_ld(lb + OFF_UH * 4, pt + CST + go);
                async_ld(lb + OFF_VH * 4, pt + 2 * CST + go);
                async_ld(lb + OFF_PH * 4, pt + 3 * CST + go);
            }
        }
    }
    wait_async0();
    __builtin_amdgcn_s_wait_tensorcnt(0);
    __syncthreads();

    // ---- Phase 2: stencil from LDS ----
    const float bulk  = fzeta + feta * (1.0f / 3.0f);
    const float c2dx2 = 2.0f * fdx * fdx;
    const float c2dtdx = 2.0f * fdt * fdx;
    const float c4dte = 4.0f * fdt * feta;
    const float c2dtb = 2.0f * fdt * bulk;

    const int ty  = tid & 31;             // y within tile
    const int xl0 = (tid >> 5) * 4;       // 4 consecutive x per thread
    const int gy  = y_base + ty;
    const int hy  = ty + 2;

    const size_t ONXY = 126ull * 126ull;
    float* out_mass = out + (((size_t)b * 2 + 0) * 62 + (size_t)(t - 1)) * ONXY;
    float* out_mom  = out + (((size_t)b * 2 + 1) * 62 + (size_t)(t - 1)) * ONXY;

#define RH_(i, j) smem[OFF_RH + (i) * HXY + (j)]
#define UH_(i, j) smem[OFF_UH + (i) * HXY + (j)]
#define VH_(i, j) smem[OFF_VH + (i) * HXY + (j)]
#define PH_(i, j) smem[OFF_PH + (i) * HXY + (j)]
#define CT_(c, i, j) smem[OFF_CT + (c) * NCTR + (i) * TY + (j)]

    if (gy < 127) {
#pragma unroll
        for (int ii = 0; ii < 4; ++ii) {
            const int xl = xl0 + ii;
            const int gx = x_base + xl;
            const int hx = xl + 2;

            const float rc   = RH_(hx, hy);
            const float r_xp = RH_(hx + 1, hy), r_xm = RH_(hx - 1, hy);
            const float r_yp = RH_(hx, hy + 1), r_ym = RH_(hx, hy - 1);

            const float uc    = UH_(hx, hy);
            const float u_xp  = UH_(hx + 1, hy), u_xm  = UH_(hx - 1, hy);
            const float u_xp2 = UH_(hx + 2, hy), u_xm2 = UH_(hx - 2, hy);
            const float u_yp  = UH_(hx, hy + 1), u_ym  = UH_(hx, hy - 1);
            const float u_pp  = UH_(hx + 1, hy + 1), u_pm = UH_(hx + 1, hy - 1);
            const float u_mp  = UH_(hx - 1, hy + 1), u_mm = UH_(hx - 1, hy - 1);

            const float vc    = VH_(hx, hy);
            const float v_xp  = VH_(hx + 1, hy), v_xm  = VH_(hx - 1, hy);
            const float v_yp  = VH_(hx, hy + 1), v_ym  = VH_(hx, hy - 1);
            const float v_yp2 = VH_(hx, hy + 2), v_ym2 = VH_(hx, hy - 2);
            const float v_pp  = VH_(hx + 1, hy + 1), v_pm = VH_(hx + 1, hy - 1);
            const float v_mp  = VH_(hx - 1, hy + 1), v_mm = VH_(hx - 1, hy - 1);

            const float p_xp = PH_(hx + 1, hy), p_xm = PH_(hx - 1, hy);
            const float p_yp = PH_(hx, hy + 1), p_ym = PH_(hx, hy - 1);

            const float r_tm = CT_(0, xl, ty), u_tm = CT_(1, xl, ty), v_tm = CT_(2, xl, ty);
            const float r_tp = CT_(3, xl, ty), u_tp = CT_(4, xl, ty), v_tp = CT_(5, xl, ty);

            const float dxu = u_xp - u_xm, dyu = u_yp - u_ym;
            const float dxv = v_xp - v_xm, dyv = v_yp - v_ym;
            const float divc = dxu + dyv;

            const float mass = (r_tp - r_tm) * fdx
                             + (rc * divc + uc * (r_xp - r_xm) + vc * (r_yp - r_ym)) * fdt;

            // D_x(div) = u_xx(2h) + cross(v);  D_y(div) = cross(u) + v_yy(2h)
            const float dxdiv = (u_xp2 - 2.0f * uc + u_xm2) + (v_pp - v_pm - v_mp + v_mm);
            const float dydiv = (u_pp - u_mp - u_pm + u_mm) + (v_yp2 - 2.0f * vc + v_ym2);

            const float lap_u = u_xp + u_xm + u_yp + u_ym - 4.0f * uc;
            const float lap_v = v_xp + v_xm + v_yp + v_ym - 4.0f * vc;

            const float dxp_ = p_xp - p_xm, dyp_ = p_yp - p_ym;
            const float dtu = u_tp - u_tm, dtv = v_tp - v_tm;

            const float mom_x = rc * dtu * c2dx2
                              + (uc * dxu + vc * dyu + dxp_) * c2dtdx
                              - lap_u * c4dte - dxdiv * c2dtb;
            const float mom_y = rc * dtv * c2dx2
                              + (uc * dxv + vc * dyv + dyp_) * c2dtdx
                              - lap_v * c4dte - dydiv * c2dtb;
            const float mom = mom_x + mom_y;

            if (gx < 127) {
                const size_t o = (size_t)(gx - 1) * 126 + (size_t)(gy - 1);
                out_mass[o] = mass;
                out_mom[o]  = mom;
            }
        }
    }
#undef RH_
#undef UH_
#undef VH_
#undef PH_
#undef CT_
}

extern "C" void kernel_launch(void* const* d_in, const int* in_sizes, int n_in,
                              void* d_out, int out_size, void* d_ws, size_t ws_size,
                              hipStream_t stream) {
    (void)in_sizes; (void)n_in; (void)out_size; (void)d_ws; (void)ws_size;
    const float* vars = (const float*)d_in[0];
    const float* dx   = (const float*)d_in[1];
    const float* dt   = (const float*)d_in[2];
    const float* eta  = (const float*)d_in[3];
    const float* zeta = (const float*)d_in[4];
    float* out = (float*)d_out;

    dim3 grid(16 /*xy tiles*/, 62 /*t'*/, 8 /*b*/);
    cns_residual_kernel<<<grid, NT, 0, stream>>>(vars, dx, dt, eta, zeta, out);
}